// RadarPillarFeatureNet_49855980372233
// MI455X (gfx1250) — compile-verified
//
#include <hip/hip_runtime.h>
#include <hip/hip_bf16.h>

typedef float v2f __attribute__((ext_vector_type(2)));
typedef float v8f __attribute__((ext_vector_type(8)));

#define NP_TOT   40000
#define MPTS     32
#define CIN      7
#define COUT     64
#define STRD     18              // LDS row stride in floats: 72B -> 8B aligned, bank-spread
#define STATS_BLOCKS 500         // 500*8 = 4000 waves -> exactly 10 pillars each (uniform)
#define MAIN_BLOCKS  5000        // 5000*8 = 40000 waves -> exactly 1 pillar each

// ---------------------------------------------------------------- wave32 sum
__device__ __forceinline__ float wave_sum32(float v) {
#pragma unroll
  for (int off = 16; off > 0; off >>= 1) v += __shfl_xor(v, off, 32);
  return v;
}

// ------------------------------------------------- load W as WMMA B operands
// B tile (4x16, f32, 2 VGPRs): v0 = rows K,K+2 striped over half-waves; v1 = K+1,K+3.
// bw[j][kc] covers cols 16j..16j+15, K = 4*kc..4*kc+3 of the 16x64 weight.
__device__ __forceinline__ void load_B(const float* __restrict__ W, int lane,
                                       v2f bw[4][4]) {
  const int lo = lane & 15;
  const int hi = lane >> 4;
#pragma unroll
  for (int j = 0; j < 4; ++j) {
#pragma unroll
    for (int kc = 0; kc < 4; ++kc) {
      const int k0 = kc * 4 + hi * 2;
      const int d  = j * 16 + lo;
      v2f b;
      b.x = W[(size_t)k0 * COUT + d];
      b.y = W[(size_t)(k0 + 1) * COUT + d];
      bw[j][kc] = b;
    }
  }
}

// -------------------- one pillar: augment -> mask -> LDS stage -> 32x WMMA --
// acc[t][j]: D tile for rows 16t..16t+15 (points), cols 16j..16j+15 (channels).
__device__ __forceinline__ void pillar_wmma(const float* __restrict__ features,
                                            const int* __restrict__ num_points,
                                            const int* __restrict__ coors,
                                            const v2f bw[4][4],
                                            float* lds_wave, int n, int lane,
                                            v8f acc[2][4]) {
  const float* fp = features + ((size_t)n * MPTS + lane) * CIN;
  const float f0 = fp[0], f1 = fp[1], f2 = fp[2], f3 = fp[3];
  const float f4 = fp[4], f5 = fp[5], f6 = fp[6];

  const int   np  = num_points[n];
  const float inv = 1.0f / (float)np;
  // means over ALL 32 points (reference sums full axis then divides by np)
  const float mx = wave_sum32(f0) * inv;
  const float my = wave_sum32(f1) * inv;
  const float mz = wave_sum32(f2) * inv;
  const float m3 = wave_sum32(f3) * inv;
  const float m4 = wave_sum32(f4) * inv;
  const float m5 = wave_sum32(f5) * inv;
  const float m6 = wave_sum32(f6) * inv;

  const float cx = (float)coors[n * 4 + 3] * 0.2f + 0.1f;            // vx/2 + 0
  const float cy = (float)coors[n * 4 + 2] * 0.2f + (0.1f - 40.0f);  // vy/2 - 40

  const float msk = (lane < np) ? 1.0f : 0.0f;

  // 16 channels: [fcx, fcy, z, v0..v3 | cluster xyz | fcx, fcy | vel-centered]
  float ch[16] = {f0 - cx, f1 - cy, f2,      f3,      f4,      f5,      f6,
                  f0 - mx, f1 - my, f2 - mz, f0 - cx, f1 - cy,
                  f3 - m3, f4 - m4, f5 - m5, f6 - m6};

  float* r = lds_wave + lane * STRD;
#pragma unroll
  for (int c = 0; c < 16; c += 2) {
    v2f p;
    p.x = ch[c] * msk;
    p.y = ch[c + 1] * msk;
    *(v2f*)(r + c) = p;  // 8B-aligned ds_store_b64
  }
  __syncthreads();  // uniform across block in both callers

  // A (16x4 f32): lane L -> row M=L%16; v0 holds K (lo half) / K+2 (hi half), v1 = +1.
#pragma unroll
  for (int t = 0; t < 2; ++t) {
#pragma unroll
    for (int kc = 0; kc < 4; ++kc) {
      const v2f a = *(const v2f*)(lds_wave + (t * 16 + (lane & 15)) * STRD +
                                  kc * 4 + ((lane >> 4) << 1));
#pragma unroll
      for (int j = 0; j < 4; ++j) {
        acc[t][j] = __builtin_amdgcn_wmma_f32_16x16x4_f32(
            false, a, false, bw[j][kc], (short)0, acc[t][j], false, false);
      }
    }
  }
}

// ------------------------------------------------------------- ws zero-init
__global__ void RPFN_zero_k(float* __restrict__ ws) {
  ws[threadIdx.x] = 0.0f;  // 128 threads: sum[64] + sumsq[64]
}

// --------------------------------------- pass 1: per-channel sum / sum-of-sq
__global__ __launch_bounds__(256) void RPFN_stats_k(
    const float* __restrict__ features, const int* __restrict__ num_points,
    const int* __restrict__ coors, const float* __restrict__ W,
    float* __restrict__ ws) {
  __shared__ float smem[8 * MPTS * STRD];
  const int lane = threadIdx.x & 31;
  const int wid  = threadIdx.x >> 5;
  float* lds_wave = smem + wid * MPTS * STRD;

  v2f bw[4][4];
  load_B(W, lane, bw);

  float ssum[4] = {0.f, 0.f, 0.f, 0.f};
  float ssq[4]  = {0.f, 0.f, 0.f, 0.f};

  const int gw = blockIdx.x * 8 + wid;
  for (int n = gw; n < NP_TOT; n += STATS_BLOCKS * 8) {  // uniform trip count
    v8f acc[2][4] = {};
    pillar_wmma(features, num_points, coors, bw, lds_wave, n, lane, acc);
#pragma unroll
    for (int j = 0; j < 4; ++j) {
      float s = 0.f, q = 0.f;
#pragma unroll
      for (int t = 0; t < 2; ++t)
#pragma unroll
        for (int v = 0; v < 8; ++v) {
          const float x = acc[t][j][v];
          s += x;
          q += x * x;
        }
      ssum[j] += s;
      ssq[j] += q;
    }
  }
  // fold lane L and L+16 (the two M-halves of each D tile)
#pragma unroll
  for (int j = 0; j < 4; ++j) {
    ssum[j] += __shfl_xor(ssum[j], 16, 32);
    ssq[j]  += __shfl_xor(ssq[j], 16, 32);
  }
  if (lane < 16) {
#pragma unroll
    for (int j = 0; j < 4; ++j) {
      __hip_atomic_fetch_add(&ws[j * 16 + lane], ssum[j], __ATOMIC_RELAXED,
                             __HIP_MEMORY_SCOPE_AGENT);
      __hip_atomic_fetch_add(&ws[64 + j * 16 + lane], ssq[j], __ATOMIC_RELAXED,
                             __HIP_MEMORY_SCOPE_AGENT);
    }
  }
}

// ----------------------------- pass 2: fold BN into per-channel scale / bias
__global__ void RPFN_finalize_k(float* __restrict__ ws,
                                const float* __restrict__ gamma,
                                const float* __restrict__ beta) {
  const int d = threadIdx.x;  // 64 threads
  const float invNM = 1.0f / (float)(NP_TOT * MPTS);
  const float mean = ws[d] * invNM;
  const float var  = ws[64 + d] * invNM - mean * mean;
  const float sc   = gamma[d] * rsqrtf(var + 1e-3f);
  ws[128 + d] = sc;
  ws[192 + d] = beta[d] - mean * sc;
}

// ------------------- pass 3: GEMM + BN + ReLU + max over points, one wave/pillar
__global__ __launch_bounds__(256) void RPFN_main_k(
    const float* __restrict__ features, const int* __restrict__ num_points,
    const int* __restrict__ coors, const float* __restrict__ W,
    const float* __restrict__ ws, float* __restrict__ out) {
  __shared__ float smem[8 * MPTS * STRD];
  const int lane = threadIdx.x & 31;
  const int wid  = threadIdx.x >> 5;
  float* lds_wave = smem + wid * MPTS * STRD;

  v2f bw[4][4];
  load_B(W, lane, bw);

  const int n = blockIdx.x * 8 + wid;  // 5000*8 == 40000 exactly
  v8f acc[2][4] = {};
  pillar_wmma(features, num_points, coors, bw, lds_wave, n, lane, acc);

  const int lo = lane & 15;
#pragma unroll
  for (int j = 0; j < 4; ++j) {
    const float sc = ws[128 + j * 16 + lo];
    const float bi = ws[192 + j * 16 + lo];
    float m = 0.0f;  // post-ReLU values are >= 0
#pragma unroll
    for (int t = 0; t < 2; ++t)
#pragma unroll
      for (int v = 0; v < 8; ++v) {
        float y = fmaf(acc[t][j][v], sc, bi);
        y = fmaxf(y, 0.0f);
        m = fmaxf(m, y);
      }
    m = fmaxf(m, __shfl_xor(m, 16, 32));
    if (lane < 16) out[(size_t)n * COUT + j * 16 + lo] = m;
  }
}

extern "C" void kernel_launch(void* const* d_in, const int* in_sizes, int n_in,
                              void* d_out, int out_size, void* d_ws,
                              size_t ws_size, hipStream_t stream) {
  (void)in_sizes; (void)n_in; (void)out_size; (void)ws_size;
  const float* features   = (const float*)d_in[0];
  const int*   num_points = (const int*)d_in[1];
  const int*   coors      = (const int*)d_in[2];
  const float* W          = (const float*)d_in[3];
  const float* gamma      = (const float*)d_in[4];
  const float* beta       = (const float*)d_in[5];
  float*       out        = (float*)d_out;
  float*       ws         = (float*)d_ws;  // [0:64) sum, [64:128) sumsq, [128:192) scale, [192:256) bias

  RPFN_zero_k<<<1, 128, 0, stream>>>(ws);
  RPFN_stats_k<<<STATS_BLOCKS, 256, 0, stream>>>(features, num_points, coors, W, ws);
  RPFN_finalize_k<<<1, 64, 0, stream>>>(ws, gamma, beta);
  RPFN_main_k<<<MAIN_BLOCKS, 256, 0, stream>>>(features, num_points, coors, W, ws, out);
}